// EigenBiasAttention_77730318123130
// MI455X (gfx1250) — compile-verified
//
#include <hip/hip_runtime.h>
#include <hip/hip_bf16.h>
#include <math.h>

// ---------------------------------------------------------------------------
// CDNA5 (gfx1250) f32 WMMA: D(16x16) = A(16x4) * B(4x16) + C, wave32.
// A layout: lane L: m = L%16, K = 2*(L/16) + r  (r = reg 0..1)
// B layout (mirrored): lane L: n = L%16, K = 2*(L/16) + r
// C/D layout: lane L: n = L%16, m = r + 8*(L/16) (r = reg 0..7)
// ---------------------------------------------------------------------------
typedef __attribute__((ext_vector_type(2))) float v2f;
typedef __attribute__((ext_vector_type(8))) float v8f;

__device__ __forceinline__ v8f wmma_f32x4(v2f a, v2f b, v8f c) {
  return __builtin_amdgcn_wmma_f32_16x16x4_f32(
      /*neg_a=*/false, a, /*neg_b=*/false, b,
      /*c_mod=*/(short)0, c, /*reuse_a=*/false, /*reuse_b=*/false);
}

// ---------------------------------------------------------------------------
// CDNA5 async direct-to-LDS copy (ASYNCcnt-tracked); bypasses VGPRs.
// LDS byte address = low 32 bits of the flat shared pointer (ISA: LDS aperture
// maps addr[31:0] directly to the wave's LDS allocation).
// ---------------------------------------------------------------------------
__device__ __forceinline__ uint32_t lds_off(const void* p) {
  return (uint32_t)(uintptr_t)p;
}
__device__ __forceinline__ void async_b128(uint32_t dst_lds, const void* src) {
  asm volatile("global_load_async_to_lds_b128 %0, %1, off"
               :: "v"(dst_lds), "v"(src) : "memory");
}
__device__ __forceinline__ void wait_async0() {
  asm volatile("s_wait_asynccnt 0" ::: "memory");
}

// ---------------------------------------------------------------------------
// Generic GEMM: C[M,N] = X[M,K] @ W[N,K]^T + bias[N]
// Block tile 128(M) x 64(N), 8 waves, each wave owns 16(M) x 64(N).
// K staged through double-buffered LDS (chunks of 32) via async-to-LDS.
// ---------------------------------------------------------------------------
#define GBM 128
#define GBN 64
#define GKC 32

__global__ __launch_bounds__(256) void gemm_xwt_kernel(
    const float* __restrict__ X, const float* __restrict__ W,
    const float* __restrict__ bias, float* __restrict__ C,
    int M, int N, int K) {
  __shared__ float sX[2][GBM][GKC];  // 2 x 16 KB
  __shared__ float sW[2][GBN][GKC];  // 2 x 8 KB

  const int tid = threadIdx.x;
  const int wave = tid >> 5, lane = tid & 31;
  const int half = lane >> 4, l16 = lane & 15;
  const int m_base = blockIdx.x * GBM;
  const int n_base = blockIdx.y * GBN;

  auto stage = [&](int bf, int k0) {
    for (int i = tid; i < GBM * (GKC / 4); i += 256) {
      int r = i / (GKC / 4), c4 = (i % (GKC / 4)) * 4;
      async_b128(lds_off(&sX[bf][r][c4]),
                 &X[(size_t)(m_base + r) * K + k0 + c4]);
    }
    for (int i = tid; i < GBN * (GKC / 4); i += 256) {
      int r = i / (GKC / 4), c4 = (i % (GKC / 4)) * 4;
      async_b128(lds_off(&sW[bf][r][c4]),
                 &W[(size_t)(n_base + r) * K + k0 + c4]);
    }
  };

  v8f acc[4] = {v8f{}, v8f{}, v8f{}, v8f{}};

  stage(0, 0);
  wait_async0();
  __syncthreads();

  int buf = 0;
  for (int k0 = 0; k0 < K; k0 += GKC) {
    // prefetch next chunk into the other buffer while we compute this one
    if (k0 + GKC < K) stage(buf ^ 1, k0 + GKC);
#pragma unroll
    for (int kk = 0; kk < GKC; kk += 4) {
      v2f a = *(v2f*)&sX[buf][wave * 16 + l16][kk + 2 * half];
#pragma unroll
      for (int nt = 0; nt < 4; nt++) {
        v2f b = *(v2f*)&sW[buf][nt * 16 + l16][kk + 2 * half];
        acc[nt] = wmma_f32x4(a, b, acc[nt]);
      }
    }
    wait_async0();
    __syncthreads();
    buf ^= 1;
  }

#pragma unroll
  for (int nt = 0; nt < 4; nt++) {
    int n = n_base + nt * 16 + l16;
    float bv = bias ? bias[n] : 0.0f;
#pragma unroll
    for (int r = 0; r < 8; r++) {
      int m = m_base + wave * 16 + r + 8 * half;
      C[(size_t)m * N + n] = acc[nt][r] + bv;
    }
  }
}

// ---------------------------------------------------------------------------
// Plucker lines + causal 6x6 scan, one block per (b,h).
// Inputs P1w/P2w/P1r/P2r: (B*T, 64) row-major, head h uses cols h*4..h*4+3.
// w1 uses row t-1 (x_prev); zero at t==0 of each batch.
// Outputs: Jw (B,H,T,6) and rdM[t] = rd[t] @ cumsum_{u<=t} Jw[u] Jw[u]^T.
// ---------------------------------------------------------------------------
__device__ __forceinline__ void ext_norm6(const float* p1, const float* p2, float* L) {
  L[0] = p1[0] * p2[1] - p1[1] * p2[0];
  L[1] = p1[0] * p2[2] - p1[2] * p2[0];
  L[2] = p1[0] * p2[3] - p1[3] * p2[0];
  L[3] = p1[1] * p2[2] - p1[2] * p2[1];
  L[4] = p1[1] * p2[3] - p1[3] * p2[1];
  L[5] = p1[2] * p2[3] - p1[3] * p2[2];
  float n2 = 0.f;
#pragma unroll
  for (int i = 0; i < 6; i++) n2 += L[i] * L[i];
  float inv = 1.0f / fmaxf(sqrtf(n2), 1e-12f);
#pragma unroll
  for (int i = 0; i < 6; i++) L[i] *= inv;
}

__global__ __launch_bounds__(256) void lines_scan_kernel(
    const float* __restrict__ P1w, const float* __restrict__ P2w,
    const float* __restrict__ P1r, const float* __restrict__ P2r,
    float* __restrict__ Jw_g, float* __restrict__ rdM_g,
    int B, int T, int H) {
  const int bh = blockIdx.x;
  const int b = bh / H, h = bh % H;
  const int tid = threadIdx.x;

  __shared__ float sJw[1024][6];  // 24 KB
  __shared__ float sRd[1024][6];  // 24 KB
  __shared__ float sS[32][36];    // 4.5 KB  (exclusive chunk prefix of outer sums)

  for (int t = tid; t < T; t += 256) {
    float w1[4], w2[4], r1[4], r2[4];
    size_t row = (size_t)(b * T + t) * 64 + h * 4;
    if (t == 0) {
      w1[0] = w1[1] = w1[2] = w1[3] = 0.f;
    } else {
      size_t rp = (size_t)(b * T + t - 1) * 64 + h * 4;
#pragma unroll
      for (int c = 0; c < 4; c++) w1[c] = P1w[rp + c];
    }
#pragma unroll
    for (int c = 0; c < 4; c++) {
      w2[c] = P2w[row + c];
      r1[c] = P1r[row + c];
      r2[c] = P2r[row + c];
    }
    float Lw[6], Lr[6];
    ext_norm6(w1, w2, Lw);
    ext_norm6(r1, r2, Lr);
    // Jw = write_lines @ J6  (sign/permutation)
    float jw[6] = {Lw[5], -Lw[4], Lw[3], Lw[2], -Lw[1], Lw[0]};
#pragma unroll
    for (int j = 0; j < 6; j++) {
      sJw[t][j] = jw[j];
      sRd[t][j] = Lr[j];
      Jw_g[((size_t)bh * T + t) * 6 + j] = jw[j];
    }
  }
  __syncthreads();

  // per-chunk sums of outer products (32 chunks x 36 channels)
  for (int idx = tid; idx < 32 * 36; idx += 256) {
    int c = idx / 36, ij = idx % 36, i = ij / 6, j = ij % 6;
    float s = 0.f;
    for (int u = c * 32; u < c * 32 + 32; u++) s += sJw[u][i] * sJw[u][j];
    sS[c][ij] = s;
  }
  __syncthreads();

  // serial exclusive prefix over the 32 chunks (36 channels in parallel)
  if (tid < 36) {
    float run = 0.f;
    for (int c = 0; c < 32; c++) {
      float v = sS[c][tid];
      sS[c][tid] = run;
      run += v;
    }
  }
  __syncthreads();

  // per-token M = base + within-chunk tail; rdM[t] = rd[t] @ M
  for (int t = tid; t < T; t += 256) {
    int c = t >> 5;
    float Mm[36];
#pragma unroll
    for (int ij = 0; ij < 36; ij++) Mm[ij] = sS[c][ij];
    for (int u = c * 32; u <= t; u++) {
      float ju[6];
#pragma unroll
      for (int j = 0; j < 6; j++) ju[j] = sJw[u][j];
#pragma unroll
      for (int i = 0; i < 6; i++)
#pragma unroll
        for (int j = 0; j < 6; j++) Mm[i * 6 + j] += ju[i] * ju[j];
    }
    float rdv[6];
#pragma unroll
    for (int i = 0; i < 6; i++) rdv[i] = sRd[t][i];
#pragma unroll
    for (int j = 0; j < 6; j++) {
      float s = 0.f;
#pragma unroll
      for (int i = 0; i < 6; i++) s += rdv[i] * Mm[i * 6 + j];
      rdM_g[((size_t)bh * T + t) * 6 + j] = s;
    }
  }
}

// ---------------------------------------------------------------------------
// Flash attention with Plucker bias, one block per (query-block of 128, b*h).
// qkv: (B*T, 3072): q at col h*64, k at 1024+h*64, v at 2048+h*64.
// logits = 0.125*q.k + |rdM[t].Jw[s]|*bias_scale[h], causal, online softmax.
// K/V staged via double-buffered async-to-LDS. Writes attn_out (B*T, 1024).
// ---------------------------------------------------------------------------
#define FKB 32  // key block

__global__ __launch_bounds__(256) void flash_attn_bias_kernel(
    const float* __restrict__ qkv, const float* __restrict__ Jw_g,
    const float* __restrict__ rdM_g, const float* __restrict__ bias_scale,
    float* __restrict__ attn_out, int B, int T, int H) {
  const int D3 = 3072, D = 1024;
  const int bh = blockIdx.y;
  const int b = bh / H, h = bh % H;
  const int qb = blockIdx.x;
  const int tid = threadIdx.x;
  const int wave = tid >> 5, lane = tid & 31;
  const int half = lane >> 4, l16 = lane & 15;
  const float scale = 0.125f;  // dh^-0.5, dh=64
  const float bscale = bias_scale[h];

  __shared__ float sK[2][FKB][64];    // 2 x 8 KB
  __shared__ float sV[2][FKB][64];    // 2 x 8 KB
  __shared__ float sJw[2][FKB][6];    // 2 x 0.75 KB
  __shared__ float sRdM[128][6];      // 3 KB
  __shared__ float sP[8][16][FKB];    // 16 KB (per-wave P staging)

  const int tq0 = qb * 128 + wave * 16;  // this wave's first query row

  for (int i = tid; i < 128 * 6; i += 256) {
    int t = qb * 128 + i / 6;
    sRdM[i / 6][i % 6] = rdM_g[((size_t)bh * T + t) * 6 + (i % 6)];
  }

  // Q fragments in native A layout, kept in registers (16 k-steps over dh=64)
  v2f qa[16];
  {
    size_t qrow = ((size_t)b * T + (tq0 + l16)) * D3 + h * 64;
#pragma unroll
    for (int ks = 0; ks < 16; ks++)
      qa[ks] = *(const v2f*)&qkv[qrow + ks * 4 + 2 * half];
  }

  auto stage_kv = [&](int bf, int s0) {
    for (int i = tid; i < FKB * 16; i += 256) {  // FKB*64/4 b128 items
      int srow = i / 16, c4 = (i % 16) * 4;
      size_t base = ((size_t)b * T + s0 + srow) * D3 + h * 64 + c4;
      async_b128(lds_off(&sK[bf][srow][c4]), &qkv[base + 1024]);
      async_b128(lds_off(&sV[bf][srow][c4]), &qkv[base + 2048]);
    }
    for (int i = tid; i < FKB * 6; i += 256)
      sJw[bf][i / 6][i % 6] = Jw_g[((size_t)bh * T + s0 + i / 6) * 6 + (i % 6)];
  };

  v8f acc_o[4] = {v8f{}, v8f{}, v8f{}, v8f{}};
  float m_i[8], l_i[8];
#pragma unroll
  for (int r = 0; r < 8; r++) { m_i[r] = -3.0e38f; l_i[r] = 0.f; }

  const int s_end = qb * 128 + 128;  // exclusive; uniform over the block

  stage_kv(0, 0);
  wait_async0();
  __syncthreads();  // also publishes sRdM

  int buf = 0;
  for (int s0 = 0; s0 < s_end; s0 += FKB) {
    // prefetch next key block while computing this one
    if (s0 + FKB < s_end) stage_kv(buf ^ 1, s0 + FKB);

    // S = Q @ K^T : two 16x16 tiles (32 keys)
    v8f accs[2] = {v8f{}, v8f{}};
#pragma unroll
    for (int ks = 0; ks < 16; ks++) {
      v2f a = qa[ks];
#pragma unroll
      for (int nt = 0; nt < 2; nt++) {
        v2f bfr = *(v2f*)&sK[buf][nt * 16 + l16][ks * 4 + 2 * half];
        accs[nt] = wmma_f32x4(a, bfr, accs[nt]);
      }
    }

    // bias + mask + online softmax (row = r + 8*half, col = nt*16 + l16)
    float p[2][8];
#pragma unroll
    for (int r = 0; r < 8; r++) {
      int m = r + 8 * half;
      int t = tq0 + m;
      float rowmax = -3.0e38f;
#pragma unroll
      for (int nt = 0; nt < 2; nt++) {
        int s = s0 + nt * 16 + l16;
        float dotv = 0.f;
#pragma unroll
        for (int j = 0; j < 6; j++)
          dotv += sRdM[wave * 16 + m][j] * sJw[buf][nt * 16 + l16][j];
        float val = accs[nt][r] * scale + fabsf(dotv) * bscale;
        if (s > t) val = -3.0e38f;
        p[nt][r] = val;
        rowmax = fmaxf(rowmax, val);
      }
#pragma unroll
      for (int off = 1; off < 16; off <<= 1)
        rowmax = fmaxf(rowmax, __shfl_xor(rowmax, off, 32));
      float mnew = fmaxf(m_i[r], rowmax);
      float corr = __expf(m_i[r] - mnew);
      float rowsum = 0.f;
#pragma unroll
      for (int nt = 0; nt < 2; nt++) {
        float e = __expf(p[nt][r] - mnew);
        p[nt][r] = e;
        rowsum += e;
      }
#pragma unroll
      for (int off = 1; off < 16; off <<= 1)
        rowsum += __shfl_xor(rowsum, off, 32);
      l_i[r] = l_i[r] * corr + rowsum;
      m_i[r] = mnew;
#pragma unroll
      for (int nt = 0; nt < 4; nt++) acc_o[nt][r] *= corr;
    }

    // stage P (C-layout -> LDS -> A-layout) for the PV matmul
#pragma unroll
    for (int nt = 0; nt < 2; nt++)
#pragma unroll
      for (int r = 0; r < 8; r++)
        sP[wave][r + 8 * half][nt * 16 + l16] = p[nt][r];
    __syncthreads();

    // O += P(16 x 32) @ V(32 x 64)
#pragma unroll
    for (int ks = 0; ks < FKB / 4; ks++) {
      v2f a = *(v2f*)&sP[wave][l16][ks * 4 + 2 * half];
#pragma unroll
      for (int nt = 0; nt < 4; nt++) {
        v2f bfr;
        bfr.x = sV[buf][ks * 4 + 2 * half + 0][nt * 16 + l16];
        bfr.y = sV[buf][ks * 4 + 2 * half + 1][nt * 16 + l16];
        acc_o[nt] = wmma_f32x4(a, bfr, acc_o[nt]);
      }
    }

    wait_async0();
    __syncthreads();
    buf ^= 1;
  }

  // normalize and write (B*T, 1024) at head column
#pragma unroll
  for (int nt = 0; nt < 4; nt++)
#pragma unroll
    for (int r = 0; r < 8; r++) {
      int t = tq0 + r + 8 * half;
      attn_out[((size_t)b * T + t) * D + h * 64 + nt * 16 + l16] =
          acc_o[nt][r] / l_i[r];
    }
}

// ---------------------------------------------------------------------------
// Host-side orchestration
// ---------------------------------------------------------------------------
extern "C" void kernel_launch(void* const* d_in, const int* in_sizes, int n_in,
                              void* d_out, int out_size, void* d_ws, size_t ws_size,
                              hipStream_t stream) {
  const float* x       = (const float*)d_in[0];
  const float* W_qkv   = (const float*)d_in[1];
  const float* b_qkv   = (const float*)d_in[2];
  const float* W1w     = (const float*)d_in[3];
  const float* W2w     = (const float*)d_in[4];
  const float* W1r     = (const float*)d_in[5];
  const float* W2r     = (const float*)d_in[6];
  const float* bscale  = (const float*)d_in[7];
  const float* W_out   = (const float*)d_in[8];
  const float* b_out   = (const float*)d_in[9];
  float* out = (float*)d_out;

  const int B = 2, T = 1024, D = 1024, H = 16;
  const int M = B * T;

  float* ws = (float*)d_ws;
  size_t o = 0;
  float* qkv  = ws + o; o += (size_t)M * 3 * D;      // 6.3M floats
  float* p1w  = ws + o; o += (size_t)M * 64;
  float* p2w  = ws + o; o += (size_t)M * 64;
  float* p1r  = ws + o; o += (size_t)M * 64;
  float* p2r  = ws + o; o += (size_t)M * 64;
  float* Jwg  = ws + o; o += (size_t)B * H * T * 6;
  float* rdMg = ws + o; o += (size_t)B * H * T * 6;
  float* aout = ws + o; o += (size_t)M * D;

  // 1) QKV projection (2048x1024 @ 1024x3072)
  gemm_xwt_kernel<<<dim3(M / GBM, 3 * D / GBN), 256, 0, stream>>>(
      x, W_qkv, b_qkv, qkv, M, 3 * D, D);
  // 2) four Plucker line projections (N=64)
  gemm_xwt_kernel<<<dim3(M / GBM, 1), 256, 0, stream>>>(x, W1w, nullptr, p1w, M, 64, D);
  gemm_xwt_kernel<<<dim3(M / GBM, 1), 256, 0, stream>>>(x, W2w, nullptr, p2w, M, 64, D);
  gemm_xwt_kernel<<<dim3(M / GBM, 1), 256, 0, stream>>>(x, W1r, nullptr, p1r, M, 64, D);
  gemm_xwt_kernel<<<dim3(M / GBM, 1), 256, 0, stream>>>(x, W2r, nullptr, p2r, M, 64, D);
  // 3) exterior products + causal 6x6 scan
  lines_scan_kernel<<<B * H, 256, 0, stream>>>(p1w, p2w, p1r, p2r, Jwg, rdMg, B, T, H);
  // 4) fused flash attention with bias
  flash_attn_bias_kernel<<<dim3(T / 128, B * H), 256, 0, stream>>>(
      qkv, Jwg, rdMg, bscale, aout, B, T, H);
  // 5) output projection
  gemm_xwt_kernel<<<dim3(M / GBM, D / GBN), 256, 0, stream>>>(
      aout, W_out, b_out, out, M, D, D);
}